// Model_54949811585357
// MI455X (gfx1250) — compile-verified
//
#include <hip/hip_runtime.h>
#include <hip/hip_bf16.h>

// ---------------------------------------------------------------------------
// 2-layer GraphSAGE (mean aggr) + edge-dot classifier for MI455X (gfx1250).
//
// Sizing:  N=50000 nodes, D=128 hidden, E=800000 edges, L=200000 label edges.
// h (25.6MB) + agg (25.6MB) + deg fit in the 192MB L2, so the random
// gather/atomic-scatter (the real cost) runs at L2 rate.  The four
// [50000,128]x[128,128] GEMMs are tiny (6.6 GFLOP total) -> keep fp32
// numerics and run them through V_WMMA_F32_16X16X4_F32.
// ---------------------------------------------------------------------------

typedef __attribute__((ext_vector_type(2))) float v2f;
typedef __attribute__((ext_vector_type(8))) float v8f;

#define HID 128
#define EMBD 64

// ---------------- utility: zero fill ----------------
__global__ void fill0_kernel(float* __restrict__ p, long long n) {
    long long i = (long long)blockIdx.x * blockDim.x + threadIdx.x;
    if (i < n) p[i] = 0.0f;
}

// ---------------- h0 = concat(emb[n_id], x_struct) ----------------
__global__ void build_h0_kernel(const int* __restrict__ n_id,
                                const float* __restrict__ emb,
                                const float* __restrict__ x_struct,
                                float* __restrict__ h0, int n_nodes) {
    long long t = (long long)blockIdx.x * blockDim.x + threadIdx.x;
    int i = (int)(t >> 7);       // node
    int c = (int)(t & 127);      // channel
    if (i >= n_nodes) return;
    float v;
    if (c < EMBD) v = emb[(size_t)n_id[i] * EMBD + c];
    else          v = x_struct[(size_t)i * EMBD + (c - EMBD)];
    h0[(size_t)i * HID + c] = v;
}

// ---------------- degree accumulation (once; same for both layers) --------
__global__ void deg_kernel(const int* __restrict__ dst, float* __restrict__ deg,
                           int n_edges) {
    int e = blockIdx.x * blockDim.x + threadIdx.x;
    if (e >= n_edges) return;
    atomicAdd(&deg[dst[e]], 1.0f);
}

__global__ void invdeg_kernel(const float* __restrict__ deg,
                              float* __restrict__ invdeg, int n_nodes) {
    int i = blockIdx.x * blockDim.x + threadIdx.x;
    if (i >= n_nodes) return;
    invdeg[i] = 1.0f / fmaxf(deg[i], 1.0f);   // clip(deg, 1.0)
}

// ---------------- scatter-add of messages: agg[dst] += h[src] ------------
// One thread per (edge, channel); consecutive threads share an edge ->
// coalesced loads from h[src] and coalesced atomics into agg[dst] (L2-resident).
__global__ void scatter_kernel(const float* __restrict__ h,
                               const int* __restrict__ src,
                               const int* __restrict__ dst,
                               float* __restrict__ agg, int n_edges) {
    long long t = (long long)blockIdx.x * blockDim.x + threadIdx.x;
    int e = (int)(t >> 7);
    int c = (int)(t & 127);
    if (e >= n_edges) return;
    int s = src[e], d = dst[e];
    atomicAdd(&agg[(size_t)d * HID + c], h[(size_t)s * HID + c]);
}

// ---------------- SAGE linear: out = (agg*invdeg)@Wl + h@Wr + b ----------
// Block = 256 threads = 8 wave32 waves; block owns a 16-row M tile,
// wave w owns columns [16w, 16w+16).  K=128 walked in steps of 4 with
// V_WMMA_F32_16X16X4_F32, two accumulators (mean path + self path).
// EXEC is all-1s for every lane around the WMMAs (no divergence).
__global__ __launch_bounds__(256)
void sage_gemm_kernel(const float* __restrict__ agg,
                      const float* __restrict__ invdeg,
                      const float* __restrict__ h,
                      const float* __restrict__ Wl,
                      const float* __restrict__ Wr,
                      const float* __restrict__ bias,
                      float* __restrict__ out,
                      int n_nodes, int do_relu) {
    const int lane = threadIdx.x & 31;
    const int wave = threadIdx.x >> 5;
    const int m0   = blockIdx.x * 16;
    const int n0   = wave * 16;

    const int lrow  = lane & 15;          // row (A) / col (B) within tile
    const int khalf = (lane >> 4) * 2;    // 0 for lanes 0-15, 2 for 16-31

    int mrow = m0 + lrow;
    if (mrow >= n_nodes) mrow = n_nodes - 1;      // clamp (keeps EXEC full)
    const float idg = invdeg[mrow];
    const float* __restrict__ hrow = h   + (size_t)mrow * HID;
    const float* __restrict__ arow = agg + (size_t)mrow * HID;

    const int ncol = n0 + lrow;

    v8f acc_l = {};   // mean-aggregate path
    v8f acc_r = {};   // self path

    for (int k0 = 0; k0 < HID; k0 += 4) {
        const int k = k0 + khalf;
        v2f a_l, a_r, b_l, b_r;
        // A fragments: lane L<16 -> A[m=L][k0..k0+1], L>=16 -> [k0+2..k0+3]
        a_l.x = arow[k]     * idg;
        a_l.y = arow[k + 1] * idg;
        a_r.x = hrow[k];
        a_r.y = hrow[k + 1];
        // B fragments (mirrored): B[k][n], lane indexes n, vgpr+half index k
        b_l.x = Wl[(size_t)k * HID + ncol];
        b_l.y = Wl[(size_t)(k + 1) * HID + ncol];
        b_r.x = Wr[(size_t)k * HID + ncol];
        b_r.y = Wr[(size_t)(k + 1) * HID + ncol];

        acc_l = __builtin_amdgcn_wmma_f32_16x16x4_f32(
            false, a_l, false, b_l, (short)0, acc_l, false, false);
        acc_r = __builtin_amdgcn_wmma_f32_16x16x4_f32(
            false, a_r, false, b_r, (short)0, acc_r, false, false);
    }

    // D layout: VGPR v holds M = m0 + v + 8*(lane>>4), N = n0 + (lane&15)
    const float bv = bias[ncol];
    const int mbase = m0 + (lane >> 4) * 8;
#pragma unroll
    for (int v = 0; v < 8; ++v) {
        const int m = mbase + v;
        if (m < n_nodes) {
            float o = acc_l[v] + acc_r[v] + bv;
            if (do_relu) o = o > 0.0f ? o : 0.0f;
            out[(size_t)m * HID + ncol] = o;
        }
    }
}

// ---------------- classifier: out[e] = h2[src_e] . h2[dst_e] -------------
__global__ void edge_dot_kernel(const float* __restrict__ h2,
                                const int* __restrict__ es,
                                const int* __restrict__ ed,
                                float* __restrict__ out, int n) {
    int e = blockIdx.x * blockDim.x + threadIdx.x;
    if (e >= n) return;
    const float4* a = (const float4*)(h2 + (size_t)es[e] * HID);
    const float4* b = (const float4*)(h2 + (size_t)ed[e] * HID);
    float s = 0.0f;
#pragma unroll 8
    for (int i = 0; i < HID / 4; ++i) {
        float4 x = a[i], y = b[i];
        s += x.x * y.x + x.y * y.y + x.z * y.z + x.w * y.w;
    }
    out[e] = s;
}

// ---------------------------------------------------------------------------
extern "C" void kernel_launch(void* const* d_in, const int* in_sizes, int n_in,
                              void* d_out, int out_size, void* d_ws, size_t ws_size,
                              hipStream_t stream) {
    const int*   n_id     = (const int*)  d_in[0];
    const float* x_struct = (const float*)d_in[1];
    const int*   eidx     = (const int*)  d_in[2];   // [2, E]
    const int*   elab     = (const int*)  d_in[3];   // [2, L]
    const float* emb      = (const float*)d_in[4];
    const float* W1l      = (const float*)d_in[5];
    const float* W1r      = (const float*)d_in[6];
    const float* b1       = (const float*)d_in[7];
    const float* W2l      = (const float*)d_in[8];
    const float* W2r      = (const float*)d_in[9];
    const float* b2       = (const float*)d_in[10];
    float*       out      = (float*)d_out;

    const int N = in_sizes[0];          // 50000
    const int E = in_sizes[2] / 2;      // 800000
    const int L = in_sizes[3] / 2;      // 200000

    const int* src = eidx;
    const int* dst = eidx + E;
    const int* lsrc = elab;
    const int* ldst = elab + L;

    // workspace layout (floats)
    float* ws     = (float*)d_ws;
    float* h0     = ws;                          // N*128
    float* h1     = h0  + (size_t)N * HID;       // N*128
    float* h2     = h1  + (size_t)N * HID;       // N*128
    float* agg    = h2  + (size_t)N * HID;       // N*128
    float* deg    = agg + (size_t)N * HID;       // N
    float* invdeg = deg + N;                     // N

    const long long nh = (long long)N * HID;

    // ---- init ----
    fill0_kernel<<<(int)((nh + 255) / 256), 256, 0, stream>>>(agg, nh);
    fill0_kernel<<<(N + 255) / 256, 256, 0, stream>>>(deg, N);
    build_h0_kernel<<<(int)((nh + 255) / 256), 256, 0, stream>>>(
        n_id, emb, x_struct, h0, N);
    deg_kernel<<<(E + 255) / 256, 256, 0, stream>>>(dst, deg, E);
    invdeg_kernel<<<(N + 255) / 256, 256, 0, stream>>>(deg, invdeg, N);

    const long long ethreads = (long long)E * HID;
    const int eblocks = (int)((ethreads + 255) / 256);
    const int mblocks = (N + 15) / 16;

    // ---- layer 1 ----
    scatter_kernel<<<eblocks, 256, 0, stream>>>(h0, src, dst, agg, E);
    sage_gemm_kernel<<<mblocks, 256, 0, stream>>>(
        agg, invdeg, h0, W1l, W1r, b1, h1, N, /*relu=*/1);

    // ---- layer 2 ----
    fill0_kernel<<<(int)((nh + 255) / 256), 256, 0, stream>>>(agg, nh);
    scatter_kernel<<<eblocks, 256, 0, stream>>>(h1, src, dst, agg, E);
    sage_gemm_kernel<<<mblocks, 256, 0, stream>>>(
        agg, invdeg, h1, W2l, W2r, b2, h2, N, /*relu=*/0);

    // ---- classifier ----
    edge_dot_kernel<<<(L + 255) / 256, 256, 0, stream>>>(h2, lsrc, ldst, out, L);
}